// AttentionSelectContext_4269197492827
// MI455X (gfx1250) — compile-verified
//
#include <hip/hip_runtime.h>
#include <math.h>

// MI455X / gfx1250: wave32, WMMA (not MFMA). Problem is memory-bound
// (~1.1 GB streamed vs ~3.2 GFLOP -> ~47 us floor at 23.3 TB/s), so:
//  - keep f32 end-to-end and use V_WMMA_F32_16X16X4_F32 for all GEMM work
//  - stream rel[] through GLOBAL_LOAD_ASYNC_TO_LDS with a depth-2 ASYNCcnt
//    pipeline (overlap score compute with the incoming stream)
//  - mark the once-read tail[] stream non-temporal so it doesn't thrash L2.

typedef __attribute__((ext_vector_type(2))) float v2f;
typedef __attribute__((ext_vector_type(8))) float v8f;

#define DDIM 256
#define MDIM 64
#define BTOT 4096
#define LN_EPS 1e-5f

#define A_STRIDE 260          // 256 + 4 pad: row r -> bank offset 4r, conflict-free column reads
#define B_STRIDE 17           // padded 16-wide B panel row
#define KCHUNK 64             // K staging chunk (keeps LDS < 64 KB conservatively)
#define BPANEL (KCHUNK * B_STRIDE + 4)  // +4 floats so the 4 wave panels land on different banks

#define RCH 16                // rel rows per async chunk (16 x 256 f32 = 16 KB)

// ---------------------------------------------------------------------------
// Kernel 1: hW = (head_right - head_left) @ bilinear_w    [4096x256]@[256x256]
// One 16x16 output tile per wave, 4 waves/block -> block covers 16 rows x 64 cols.
// ---------------------------------------------------------------------------
__global__ __launch_bounds__(128) void hw_gemm_kernel(
    const float* __restrict__ head_left, const float* __restrict__ head_right,
    const float* __restrict__ bilinear_w, float* __restrict__ hW)
{
  __shared__ float sA[16 * A_STRIDE];   // weak_rel tile 16 x 256
  __shared__ float sB[4 * BPANEL];      // per-wave B panel: KCHUNK(K) x 16(N)

  const int b0   = blockIdx.x * 16;
  const int nblk = blockIdx.y * 64;
  const int tid  = threadIdx.x;
  const int wave = tid >> 5;
  const int lane = tid & 31;
  const int mrow = lane & 15;
  const int khi  = (lane >> 4) << 1;     // 0 or 2 (A/B K sub-offset per half-wave)
  const int mbase = (lane >> 4) << 3;    // D rows r / r+8

  // stage A = head_right - head_left (full K)
  for (int idx = tid; idx < 16 * DDIM; idx += 128) {
    int r = idx >> 8, c = idx & 255;
    sA[r * A_STRIDE + c] = head_right[(b0 + r) * DDIM + c] - head_left[(b0 + r) * DDIM + c];
  }

  v8f acc = {};
  float* bp = &sB[wave * BPANEL];

  for (int kc = 0; kc < DDIM; kc += KCHUNK) {
    // stage B chunk: bilinear_w[kc+k][nblk+n] -> panel (n>>4), row k, col n&15
    for (int idx = tid; idx < KCHUNK * 64; idx += 128) {
      int k = idx >> 6, n = idx & 63;   // coalesced along n
      sB[(n >> 4) * BPANEL + k * B_STRIDE + (n & 15)] = bilinear_w[(kc + k) * DDIM + nblk + n];
    }
    __syncthreads();
    #pragma unroll 4
    for (int k0 = 0; k0 < KCHUNK; k0 += 4) {
      v2f a, b;
      a.x = sA[mrow * A_STRIDE + kc + k0 + khi];
      a.y = sA[mrow * A_STRIDE + kc + k0 + khi + 1];
      b.x = bp[(k0 + khi) * B_STRIDE + mrow];
      b.y = bp[(k0 + khi + 1) * B_STRIDE + mrow];
      acc = __builtin_amdgcn_wmma_f32_16x16x4_f32(false, a, false, b, (short)0, acc, false, false);
    }
    __syncthreads();
  }

  const int ncol = nblk + wave * 16 + (lane & 15);
  #pragma unroll
  for (int r = 0; r < 8; ++r)
    hW[(b0 + mbase + r) * DDIM + ncol] = acc[r];
}

// ---------------------------------------------------------------------------
// Kernel 2: per-b masked attention, both sides. Carries the 1.07 GB stream.
// rel arrives through a depth-2 GLOBAL_LOAD_ASYNC_TO_LDS pipeline (ASYNCcnt);
// tail is read once with TH_NT so it bypasses long-term L2 residency.
// ---------------------------------------------------------------------------
__global__ __launch_bounds__(256) void attn_kernel(
    const float* __restrict__ hW,
    const float* __restrict__ rel_left,  const float* __restrict__ tail_left,
    const unsigned char* __restrict__ mask_left,
    const float* __restrict__ rel_right, const float* __restrict__ tail_right,
    const unsigned char* __restrict__ mask_right,
    float* __restrict__ attL, float* __restrict__ attR)
{
  __shared__ float s_rel[2][RCH * DDIM];   // double buffer, 2 x 16 KB
  __shared__ float s_h[DDIM];
  __shared__ float s_sc[2][MDIM];
  __shared__ float s_att[2][MDIM];

  const int b    = blockIdx.x;
  const int tid  = threadIdx.x;
  const int wave = tid >> 5;
  const int lane = tid & 31;

  const float* relside[2]  = {rel_left, rel_right};
  const float* tailside[2] = {tail_left, tail_right};
  const unsigned char* maskside[2] = {mask_left, mask_right};

  s_h[tid] = hW[b * DDIM + tid];

  // cover the tail streams with prefetches (-> global_prefetch_b8)
  {
    const char* t0 = (const char*)(tail_left  + (size_t)b * MDIM * DDIM);
    const char* t1 = (const char*)(tail_right + (size_t)b * MDIM * DDIM);
    __builtin_prefetch(t0 + tid * 256, 0, 0);
    __builtin_prefetch(t0 + tid * 256 + 128, 0, 0);
    __builtin_prefetch(t1 + tid * 256, 0, 0);
    __builtin_prefetch(t1 + tid * 256 + 128, 0, 0);
  }

  // issue one 16-row rel chunk (16 KB) as 4 async b128 copies per thread:
  // chunk c: side c>>2, rows (c&3)*16 .. +16, into LDS buffer c&1.
  auto issue_chunk = [&](int c) {
    const float* src = relside[c >> 2] + ((size_t)b * MDIM + (size_t)(c & 3) * RCH) * DDIM;
    unsigned lds0 = (unsigned)(uintptr_t)(void*)&s_rel[c & 1][0];
    #pragma unroll
    for (int j = 0; j < 4; ++j) {
      int byteoff = (j * 256 + tid) * 16;              // float4 granularity
      unsigned loff = lds0 + (unsigned)byteoff;
      asm volatile("global_load_async_to_lds_b128 %0, %1, %2"
                   :: "v"(loff), "v"(byteoff), "s"(src) : "memory");
    }
  };

  issue_chunk(0);
  issue_chunk(1);

  // ---- score pipeline: 8 chunks (2 sides x 4), depth 2 ----
  for (int c = 0; c < 8; ++c) {
    if (c < 7) asm volatile("s_wait_asynccnt 0x4" ::: "memory");  // chunk c complete
    else       asm volatile("s_wait_asynccnt 0x0" ::: "memory");
    __syncthreads();                                   // whole chunk visible to all waves

    const int side = c >> 2;
    const float* buf = s_rel[c & 1];
    #pragma unroll
    for (int i = 0; i < 2; ++i) {                      // wave w -> chunk rows 2w, 2w+1
      int mc = 2 * wave + i;
      int mg = (c & 3) * RCH + mc;
      float p = 0.f;
      #pragma unroll
      for (int k = 0; k < 8; ++k)
        p = fmaf(s_h[lane + 32 * k], buf[mc * DDIM + lane + 32 * k], p);
      #pragma unroll
      for (int off = 16; off > 0; off >>= 1) p += __shfl_xor(p, off, 32);
      if (lane == 0)
        s_sc[side][mg] = maskside[side][b * MDIM + mg] ? -INFINITY : p;
    }
    __syncthreads();                                   // all reads of buf done
    if (c + 2 < 8) issue_chunk(c + 2);                 // refill freed buffer
  }

  // ---- masked softmax over 64, waves 0/1 handle sides 0/1 in parallel ----
  if (wave < 2) {
    float v0 = s_sc[wave][lane], v1 = s_sc[wave][lane + 32];
    float mx = fmaxf(v0, v1);
    #pragma unroll
    for (int off = 16; off > 0; off >>= 1) mx = fmaxf(mx, __shfl_xor(mx, off, 32));
    float e0 = expf(v0 - mx), e1 = expf(v1 - mx);
    float sm = e0 + e1;
    #pragma unroll
    for (int off = 16; off > 0; off >>= 1) sm += __shfl_xor(sm, off, 32);
    float inv = 1.f / sm;
    s_att[wave][lane]      = e0 * inv;
    s_att[wave][lane + 32] = e1 * inv;
  }
  __syncthreads();

  // ---- weighted tail sums (column-per-thread, NT streaming loads) ----
  float aL = 0.f, aR = 0.f;
  const float* tl = tail_left  + (size_t)b * MDIM * DDIM + tid;
  const float* tr = tail_right + (size_t)b * MDIM * DDIM + tid;
  #pragma unroll 8
  for (int m = 0; m < MDIM; ++m) {
    aL = fmaf(s_att[0][m], __builtin_nontemporal_load(tl + m * DDIM), aL);
    aR = fmaf(s_att[1][m], __builtin_nontemporal_load(tr + m * DDIM), aR);
  }
  attL[b * DDIM + tid] = aL;
  attR[b * DDIM + tid] = aR;
}

// ---------------------------------------------------------------------------
// Kernel 3: y = relu(att @ w_tail^T + head @ w_head^T) + head  (pre-LN, to d_out)
// Same WMMA tiling as kernel 1; B is transposed while staging into LDS.
// ---------------------------------------------------------------------------
__global__ __launch_bounds__(128) void out_gemm_kernel(
    const float* __restrict__ att, const float* __restrict__ head,
    const float* __restrict__ w_tail, const float* __restrict__ w_head,
    float* __restrict__ y)
{
  __shared__ float sAa[16 * A_STRIDE];  // att tile 16 x 256
  __shared__ float sAh[16 * A_STRIDE];  // head tile 16 x 256 (also residual source)
  __shared__ float sB[4 * BPANEL];      // shared B chunk, re-staged per weight/K-chunk

  const int b0   = blockIdx.x * 16;
  const int nblk = blockIdx.y * 64;
  const int tid  = threadIdx.x;
  const int wave = tid >> 5;
  const int lane = tid & 31;
  const int mrow = lane & 15;
  const int khi  = (lane >> 4) << 1;
  const int mbase = (lane >> 4) << 3;

  for (int idx = tid; idx < 16 * DDIM; idx += 128) {
    int r = idx >> 8, c = idx & 255;
    sAa[r * A_STRIDE + c] = att [(b0 + r) * DDIM + c];
    sAh[r * A_STRIDE + c] = head[(b0 + r) * DDIM + c];
  }

  v8f acc = {};
  float* bp = &sB[wave * BPANEL];

  for (int ph = 0; ph < 2; ++ph) {
    const float* W  = ph ? w_head : w_tail;
    const float* Ax = ph ? sAh    : sAa;
    for (int kc = 0; kc < DDIM; kc += KCHUNK) {
      // stage transposed weight chunk: B[k][n] = W[(nblk+n)*256 + kc+k]
      for (int idx = tid; idx < KCHUNK * 64; idx += 128) {
        int n = idx >> 6, k = idx & 63;  // coalesced along k (rows of W)
        sB[(n >> 4) * BPANEL + k * B_STRIDE + (n & 15)] = W[(nblk + n) * DDIM + kc + k];
      }
      __syncthreads();
      #pragma unroll 4
      for (int k0 = 0; k0 < KCHUNK; k0 += 4) {
        v2f a, b;
        a.x = Ax[mrow * A_STRIDE + kc + k0 + khi];
        a.y = Ax[mrow * A_STRIDE + kc + k0 + khi + 1];
        b.x = bp[(k0 + khi) * B_STRIDE + mrow];
        b.y = bp[(k0 + khi + 1) * B_STRIDE + mrow];
        acc = __builtin_amdgcn_wmma_f32_16x16x4_f32(false, a, false, b, (short)0, acc, false, false);
      }
      __syncthreads();
    }
  }

  const int ncol = nblk + wave * 16 + (lane & 15);
  #pragma unroll
  for (int r = 0; r < 8; ++r) {
    int m = mbase + r;
    float v = fmaxf(acc[r], 0.f) + sAh[m * A_STRIDE + ncol];  // relu + residual
    y[(b0 + m) * DDIM + ncol] = v;
  }
}

// ---------------------------------------------------------------------------
// Kernel 4: in-place LayerNorm over d=256, one wave per row.
// ---------------------------------------------------------------------------
__global__ __launch_bounds__(256) void ln_kernel(
    float* __restrict__ y, const float* __restrict__ gamma, const float* __restrict__ beta)
{
  const int row  = blockIdx.x * 8 + (threadIdx.x >> 5);
  const int lane = threadIdx.x & 31;
  float x[8];
  float s = 0.f, ss = 0.f;
  #pragma unroll
  for (int k = 0; k < 8; ++k) {
    x[k] = y[row * DDIM + lane + 32 * k];
    s += x[k];
    ss = fmaf(x[k], x[k], ss);
  }
  #pragma unroll
  for (int off = 16; off > 0; off >>= 1) {
    s  += __shfl_xor(s,  off, 32);
    ss += __shfl_xor(ss, off, 32);
  }
  float mu  = s * (1.f / 256.f);
  float var = ss * (1.f / 256.f) - mu * mu;
  float rs  = rsqrtf(var + LN_EPS);
  #pragma unroll
  for (int k = 0; k < 8; ++k) {
    int c = lane + 32 * k;
    y[row * DDIM + c] = (x[k] - mu) * rs * gamma[c] + beta[c];
  }
}

// ---------------------------------------------------------------------------
extern "C" void kernel_launch(void* const* d_in, const int* in_sizes, int n_in,
                              void* d_out, int out_size, void* d_ws, size_t ws_size,
                              hipStream_t stream) {
  const float* head_left  = (const float*)d_in[0];
  const float* rel_left   = (const float*)d_in[1];
  const float* tail_left  = (const float*)d_in[2];
  const float* head_right = (const float*)d_in[3];
  const float* rel_right  = (const float*)d_in[4];
  const float* tail_right = (const float*)d_in[5];
  const unsigned char* mask_left  = (const unsigned char*)d_in[6];  // numpy bool = 1 byte
  const unsigned char* mask_right = (const unsigned char*)d_in[7];
  const float* bilinear_w = (const float*)d_in[8];
  const float* w_tail     = (const float*)d_in[9];
  const float* w_head     = (const float*)d_in[10];
  const float* ln_gamma   = (const float*)d_in[11];
  const float* ln_beta    = (const float*)d_in[12];

  float* outL = (float*)d_out;
  float* outR = outL + (size_t)BTOT * DDIM;

  // workspace: hW | attL | attR  (3 x 4 MB = 12 MB), each fully written before read
  float* hW   = (float*)d_ws;
  float* attL = hW   + (size_t)BTOT * DDIM;
  float* attR = attL + (size_t)BTOT * DDIM;

  dim3 gGemm(BTOT / 16, DDIM / 64);
  hw_gemm_kernel<<<gGemm, 128, 0, stream>>>(head_left, head_right, bilinear_w, hW);
  attn_kernel<<<BTOT, 256, 0, stream>>>(hW, rel_left, tail_left, mask_left,
                                        rel_right, tail_right, mask_right, attL, attR);
  out_gemm_kernel<<<gGemm, 128, 0, stream>>>(attL, head_left,  w_tail, w_head, outL);
  out_gemm_kernel<<<gGemm, 128, 0, stream>>>(attR, head_right, w_tail, w_head, outR);
  ln_kernel<<<(2 * BTOT) / 8, 256, 0, stream>>>((float*)d_out, ln_gamma, ln_beta);
}